// AdvancedGAT_80625126080955
// MI455X (gfx1250) — compile-verified
//
#include <hip/hip_runtime.h>
#include <hip/hip_bf16.h>

typedef __attribute__((ext_vector_type(2))) float v2f;
typedef __attribute__((ext_vector_type(8))) float v8f;

#define LEAKK 0.2f
#define EPSS  1e-16f
#define BNEPS 1e-5f

__device__ inline float lrelu(float x) { return x > 0.f ? x : LEAKK * x; }
// order-preserving float->uint map for atomicMax-based segment max
__device__ inline unsigned fmapu(float f) {
    unsigned u = __float_as_uint(f);
    return (u & 0x80000000u) ? ~u : (u | 0x80000000u);
}
__device__ inline float funmap(unsigned u) {
    return (u & 0x80000000u) ? __uint_as_float(u & 0x7FFFFFFFu) : __uint_as_float(~u);
}
__device__ inline void edge_sd(const int* __restrict__ ei, int E, int e, int& s, int& d) {
    if (e < E) { s = ei[e]; d = ei[E + e]; } else { s = d = e - E; }
}

// ---------------- fills ----------------
__global__ void fill_f32(float* p, float v, int n) {
    int i = blockIdx.x * blockDim.x + threadIdx.x;
    if (i < n) p[i] = v;
}
__global__ void fill_u32(unsigned* p, unsigned v, int n) {
    int i = blockIdx.x * blockDim.x + threadIdx.x;
    if (i < n) p[i] = v;
}

// ---------------- f32 WMMA GEMM with LDS double buffering ----------------
// H = X[R,K] @ W[K,Cn], V_WMMA_F32_16X16X4_F32 (exact f32 semantics).
// Block = 256 threads = 8 waves arranged (8/WCOLS) x WCOLS; wave tile 16 x (16*NACC).
// Block tile BM x BN, K-chunk 32, double-buffered LDS; global loads for chunk c+1
// are issued before the unrolled WMMA phase of chunk c (latency hiding).
// A LDS pitch 36: m*36 mod 64 distinct for m=0..15, half-wave K+2 offset disjoint.
// B LDS uses a K-pair interleaved layout Bs[kp][col][2] = {B[2kp][col], B[2kp+1][col]}
// so one B fragment = one aligned ds_load_b64 into an even VGPR pair (no v_movs);
// pair-row pitch 2*BN+32 => half-wave kp+1 offset = 32 mod 64 banks -> conflict-free.
template <int WCOLS, int NACC, int BM>
__global__ __launch_bounds__(256) void gemm_wmma_lds(const float* __restrict__ X,
                                                     const float* __restrict__ W,
                                                     float* __restrict__ Hout,
                                                     int R, int K, int Cn) {
    constexpr int BN     = WCOLS * 16 * NACC;
    constexpr int KC     = 32;
    constexpr int KP     = KC / 2;          // pair rows
    constexpr int KCP    = 36;              // A LDS pitch (floats)
    constexpr int BPITCH = 2 * BN + 32;     // B pair-row pitch (floats)
    constexpr int AREG   = BM / 32;         // float4 per thread for the A tile
    constexpr int BWORK  = 4 * BN;          // pair-items: KP * (BN/4)
    constexpr int BI     = (BWORK + 255) / 256;

    __shared__ float As[2][BM * KCP];
    __shared__ float Bs[2][KP * BPITCH];

    const int tid  = threadIdx.x;
    const int wid  = tid >> 5;
    const int lane = tid & 31;
    const int lm   = lane & 15;
    const int lk   = (lane >> 4) * 2;          // K sub-offset: 0 or 2
    const int rowW = (wid / WCOLS) * 16;       // wave row offset in block tile
    const int colW = (wid % WCOLS) * 16 * NACC;
    const int bRow0 = blockIdx.y * BM;
    const int col0  = blockIdx.x * BN;
    const int nchunk = K / KC;

    float4 ar[AREG], br0[BI], br1[BI];

    auto loadA = [&](int kb) {
#pragma unroll
        for (int i = 0; i < AREG; ++i) {
            int idx = tid + i * 256;
            int row = idx >> 3, q = idx & 7;   // 8 float4 per 32-float row
            int gr = bRow0 + row; if (gr >= R) gr = R - 1;  // clamp (barrier-safe)
            ar[i] = *(const float4*)(X + (size_t)gr * K + kb + q * 4);
        }
    };
    auto loadB = [&](int kb) {
#pragma unroll
        for (int i = 0; i < BI; ++i) {
            int idx = tid + i * 256;
            if (BWORK % 256 == 0 || idx < BWORK) {
                int kp = idx / (BN / 4), c4 = idx % (BN / 4);
                const float* p = W + (size_t)(kb + 2 * kp) * Cn + col0 + c4 * 4;
                br0[i] = *(const float4*)p;
                br1[i] = *(const float4*)(p + Cn);
            }
        }
    };
    auto storeA = [&](int buf) {
#pragma unroll
        for (int i = 0; i < AREG; ++i) {
            int idx = tid + i * 256;
            int row = idx >> 3, q = idx & 7;
            *(float4*)&As[buf][row * KCP + q * 4] = ar[i];
        }
    };
    auto storeB = [&](int buf) {
#pragma unroll
        for (int i = 0; i < BI; ++i) {
            int idx = tid + i * 256;
            if (BWORK % 256 == 0 || idx < BWORK) {
                int kp = idx / (BN / 4), c4 = idx % (BN / 4);
                int base = kp * BPITCH + c4 * 8;
                *(float4*)&Bs[buf][base]     = make_float4(br0[i].x, br1[i].x, br0[i].y, br1[i].y);
                *(float4*)&Bs[buf][base + 4] = make_float4(br0[i].z, br1[i].z, br0[i].w, br1[i].w);
            }
        }
    };

    v8f acc[NACC] = {};

    loadA(0); loadB(0); storeA(0); storeB(0);
    __syncthreads();

    int buf = 0;
    for (int c = 0; c < nchunk; ++c) {
        if (c + 1 < nchunk) { loadA((c + 1) * KC); loadB((c + 1) * KC); }
#pragma unroll
        for (int k = 0; k < KC; k += 4) {
            v2f a = *(const v2f*)&As[buf][(rowW + lm) * KCP + k + lk];
            const int kp = (k + lk) >> 1;
#pragma unroll
            for (int j = 0; j < NACC; ++j) {
                v2f b = *(const v2f*)&Bs[buf][kp * BPITCH + (colW + j * 16 + lm) * 2];
                acc[j] = __builtin_amdgcn_wmma_f32_16x16x4_f32(false, a, false, b,
                                                               (short)0, acc[j], false, false);
            }
        }
        if (c + 1 < nchunk) { storeA(buf ^ 1); storeB(buf ^ 1); }
        __syncthreads();
        buf ^= 1;
    }

    // C/D layout: VGPR r -> M = r (lanes 0-15) / r+8 (lanes 16-31), N = lane&15
    const int rr = (lane >> 4) * 8;
#pragma unroll
    for (int j = 0; j < NACC; ++j)
#pragma unroll
        for (int r = 0; r < 8; ++r) {
            int grow = bRow0 + rowW + rr + r;
            if (grow < R)
                Hout[(size_t)grow * Cn + col0 + colW + j * 16 + lm] = acc[j][r];
        }
}

// ---------------- per-(node,head) attention scores ----------------
__global__ void scores_k(const float* __restrict__ Hf, const float* __restrict__ as_,
                         const float* __restrict__ ad_, float* __restrict__ ss,
                         float* __restrict__ sd, int NH, int H, int C) {
    int w = blockIdx.x * (blockDim.x >> 5) + (threadIdx.x >> 5);
    if (w >= NH) return;
    int lane = threadIdx.x & 31;
    const float* __restrict__ hp = Hf + (size_t)w * C;
    const int hoff = (w % H) * C;
    float s1 = 0.f, s2 = 0.f;
    for (int c = lane; c < C; c += 32) {
        float hv = hp[c];
        s1 += hv * as_[hoff + c];
        s2 += hv * ad_[hoff + c];
    }
#pragma unroll
    for (int m = 16; m > 0; m >>= 1) {
        s1 += __shfl_xor(s1, m, 32);
        s2 += __shfl_xor(s2, m, 32);
    }
    if (lane == 0) { ss[w] = s1; sd[w] = s2; }
}

// ---------------- edge pass 1: segment max over dst ----------------
__global__ void edge_max_k(const float* __restrict__ ss, const float* __restrict__ sd,
                           const int* __restrict__ ei, unsigned* __restrict__ maxb,
                           int E, int ET, int H) {
    int t = blockIdx.x * blockDim.x + threadIdx.x;
    if (t >= ET * H) return;
    int e = t / H, h = t - e * H;
    int s, d; edge_sd(ei, E, e, s, d);
    float sc = lrelu(ss[s * H + h] + sd[d * H + h]);
    atomicMax(&maxb[d * H + h], fmapu(sc));
}

// ---------------- edge pass 2: exp + segment sum ----------------
__global__ void edge_sum_k(const float* __restrict__ ss, const float* __restrict__ sd,
                           const int* __restrict__ ei, const unsigned* __restrict__ maxb,
                           float* __restrict__ sumb, float* __restrict__ alphaw,
                           int E, int ET, int H) {
    int t = blockIdx.x * blockDim.x + threadIdx.x;
    if (t >= ET * H) return;
    int e = t / H, h = t - e * H;
    int s, d; edge_sd(ei, E, e, s, d);
    float sc = lrelu(ss[s * H + h] + sd[d * H + h]);
    float w = expf(sc - funmap(maxb[d * H + h]));
    alphaw[(size_t)e * H + h] = w;
    atomicAdd(&sumb[d * H + h], w);
}

// ---------------- edge pass 3: out[dst] += h[src] * alpha (wave per edge) ----------------
__global__ void scatter_k(const float* __restrict__ Hf, const float* __restrict__ alphaw,
                          const float* __restrict__ sumb, const int* __restrict__ ei,
                          float* __restrict__ Ob, int E, int ET, int H, int HC, int shiftC) {
    int we = blockIdx.x * (blockDim.x >> 5) + (threadIdx.x >> 5);
    if (we >= ET) return;
    int lane = threadIdx.x & 31;
    int s, d; edge_sd(ei, E, we, s, d);
    const float* __restrict__ hr = Hf + (size_t)s * HC;
    float* __restrict__ orow = Ob + (size_t)d * HC;
    for (int i = lane; i < HC; i += 32) {
        int h = i >> shiftC;
        float a = alphaw[(size_t)we * H + h] / (sumb[d * H + h] + EPSS);
        atomicAdd(&orow[i], hr[i] * a);
    }
}

// ---------------- bias + BN + ReLU (in place capable) ----------------
__global__ void finalize_k(const float* __restrict__ Ob, const float* __restrict__ b,
                           const float* __restrict__ g, const float* __restrict__ be,
                           const float* __restrict__ m, const float* __restrict__ v,
                           float* __restrict__ Xn, int total, int outd) {
    int i = blockIdx.x * blockDim.x + threadIdx.x;
    if (i >= total) return;
    int j = i % outd;
    float x = Ob[i] + b[j];
    float y = (x - m[j]) * rsqrtf(v[j] + BNEPS) * g[j] + be[j];
    Xn[i] = fmaxf(y, 0.f);
}

// ---------------- head MLP: 32 -> 16 (ReLU) -> 2 -> log_softmax ----------------
__global__ void mlp_k(const float* __restrict__ X3, const float* __restrict__ Wc1,
                      const float* __restrict__ bc1, const float* __restrict__ Wc2,
                      const float* __restrict__ bc2, float* __restrict__ out, int Nn) {
    __shared__ float w1[32 * 16], b1s[16], w2[16 * 2], b2s[2];
    for (int i = threadIdx.x; i < 512; i += blockDim.x) w1[i] = Wc1[i];
    if (threadIdx.x < 16) b1s[threadIdx.x] = bc1[threadIdx.x];
    if (threadIdx.x < 32) w2[threadIdx.x] = Wc2[threadIdx.x];
    if (threadIdx.x < 2)  b2s[threadIdx.x] = bc2[threadIdx.x];
    __syncthreads();
    int n = blockIdx.x * blockDim.x + threadIdx.x;
    if (n >= Nn) return;
    const float* __restrict__ xp = X3 + (size_t)n * 32;
    float x[32];
#pragma unroll
    for (int i = 0; i < 32; ++i) x[i] = xp[i];
    float o0 = b2s[0], o1 = b2s[1];
#pragma unroll
    for (int j = 0; j < 16; ++j) {
        float acc = b1s[j];
#pragma unroll
        for (int i = 0; i < 32; ++i) acc += x[i] * w1[i * 16 + j];
        acc = fmaxf(acc, 0.f);
        o0 += acc * w2[j * 2 + 0];
        o1 += acc * w2[j * 2 + 1];
    }
    float mx = fmaxf(o0, o1);
    float lse = mx + logf(expf(o0 - mx) + expf(o1 - mx));
    out[(size_t)n * 2 + 0] = o0 - lse;
    out[(size_t)n * 2 + 1] = o1 - lse;
}

// ---------------- host: one GAT layer ----------------
static void run_gat_layer(hipStream_t stream, const float* Xin, int Nn, int E, int ET,
                          const int* ei, int K, int C, int H,
                          const float* W, const float* as_, const float* ad_,
                          const float* b, const float* g, const float* be,
                          const float* m, const float* v,
                          float* Hbuf, float* Obuf,
                          float* ss, float* sd, unsigned* maxb, float* sumb, float* alphaw) {
    const int HC = H * C;
    int shiftC = 0; while ((1 << shiftC) < C) ++shiftC;

    // GEMM: H = X @ W  (Cn = HC)
    if (HC % 128 == 0) {
        dim3 grid(HC / 128, (Nn + 63) / 64);
        gemm_wmma_lds<2, 4, 64><<<grid, 256, 0, stream>>>(Xin, W, Hbuf, Nn, K, HC);
    } else {
        dim3 grid(HC / 32, (Nn + 127) / 128);
        gemm_wmma_lds<1, 2, 128><<<grid, 256, 0, stream>>>(Xin, W, Hbuf, Nn, K, HC);
    }

    // scores
    int NH = Nn * H;
    scores_k<<<(NH + 7) / 8, 256, 0, stream>>>(Hbuf, as_, ad_, ss, sd, NH, H, C);

    // zero-init accumulators (graph-capture safe)
    fill_u32<<<(NH + 255) / 256, 256, 0, stream>>>(maxb, 0u, NH);   // 0 < mapped(-inf)
    fill_f32<<<(NH + 255) / 256, 256, 0, stream>>>(sumb, 0.f, NH);
    fill_f32<<<(Nn * HC + 255) / 256, 256, 0, stream>>>(Obuf, 0.f, Nn * HC);

    int ETH = ET * H;
    edge_max_k<<<(ETH + 255) / 256, 256, 0, stream>>>(ss, sd, ei, maxb, E, ET, H);
    edge_sum_k<<<(ETH + 255) / 256, 256, 0, stream>>>(ss, sd, ei, maxb, sumb, alphaw, E, ET, H);
    scatter_k<<<(ET + 7) / 8, 256, 0, stream>>>(Hbuf, alphaw, sumb, ei, Obuf, E, ET, H, HC, shiftC);

    // bias + BN + ReLU, in place -> Obuf becomes next layer's input
    finalize_k<<<(Nn * HC + 255) / 256, 256, 0, stream>>>(Obuf, b, g, be, m, v, Obuf, Nn * HC, HC);
}

extern "C" void kernel_launch(void* const* d_in, const int* in_sizes, int n_in,
                              void* d_out, int out_size, void* d_ws, size_t ws_size,
                              hipStream_t stream) {
    const float* x  = (const float*)d_in[0];
    const int*   ei = (const int*)d_in[1];
    const int Nn = in_sizes[0] / 128;
    const int E  = in_sizes[1] / 2;
    const int ET = E + Nn;

    // workspace carve (floats)
    float* P0 = (float*)d_ws;                 // H buffer (up to Nn*1024)
    float* P1 = P0 + (size_t)Nn * 1024;       // O layer1 -> X1 ; O layer3 -> X3
    float* P2 = P1 + (size_t)Nn * 1024;       // O layer2 -> X2 (Nn*256)
    float* ss = P2 + (size_t)Nn * 256;
    float* sd = ss + (size_t)Nn * 8;
    unsigned* maxb = (unsigned*)(sd + (size_t)Nn * 8);
    float* sumb   = (float*)(maxb + (size_t)Nn * 8);
    float* alphaw = sumb + (size_t)Nn * 8;    // ET*8 floats

    #define LW(i, off) ((const float*)d_in[2 + 8 * (i) + (off)])

    // Layer 1: din=128, C=128, H=8 -> X1 in P1
    run_gat_layer(stream, x, Nn, E, ET, ei, 128, 128, 8,
                  LW(0,0), LW(0,1), LW(0,2), LW(0,3), LW(0,4), LW(0,5), LW(0,6), LW(0,7),
                  P0, P1, ss, sd, maxb, sumb, alphaw);
    // Layer 2: din=1024, C=64, H=4 -> X2 in P2
    run_gat_layer(stream, P1, Nn, E, ET, ei, 1024, 64, 4,
                  LW(1,0), LW(1,1), LW(1,2), LW(1,3), LW(1,4), LW(1,5), LW(1,6), LW(1,7),
                  P0, P2, ss, sd, maxb, sumb, alphaw);
    // Layer 3: din=256, C=32, H=1 (concat=False but H==1 -> mean is identity) -> X3 in P1
    run_gat_layer(stream, P2, Nn, E, ET, ei, 256, 32, 1,
                  LW(2,0), LW(2,1), LW(2,2), LW(2,3), LW(2,4), LW(2,5), LW(2,6), LW(2,7),
                  P0, P1, ss, sd, maxb, sumb, alphaw);

    // head MLP + log_softmax
    mlp_k<<<(Nn + 255) / 256, 256, 0, stream>>>(P1, (const float*)d_in[26], (const float*)d_in[27],
                                                (const float*)d_in[28], (const float*)d_in[29],
                                                (float*)d_out, Nn);
    (void)n_in; (void)out_size; (void)ws_size;
}